// CommunityAwareSRLayer_35433480192085
// MI455X (gfx1250) — compile-verified
//
#include <hip/hip_runtime.h>
#include <hip/hip_bf16.h>

#define LR 1024
#define HR 2048
#define KCOMM 8
#define RANK 16
#define JSWEEPS 8

typedef float v2f __attribute__((ext_vector_type(2)));
typedef float v8f __attribute__((ext_vector_type(8)));
typedef int i32x2 __attribute__((vector_size(8)));

#define GLOBAL_AS __attribute__((address_space(1)))
#define LDS_AS __attribute__((address_space(3)))

// ---------------------------------------------------------------------------
// CDNA5 async global->LDS copies (ASYNCcnt path), with safe fallbacks.
// Signatures (from hipcc diagnostics): (T AS1*, T AS3*, imm offset, imm cpol)
// with T = int (b32), int __vector(2) (b64).
// ---------------------------------------------------------------------------
__device__ __forceinline__ void async_b64(const float* g, float2* l) {
#if __has_builtin(__builtin_amdgcn_global_load_async_to_lds_b64)
  __builtin_amdgcn_global_load_async_to_lds_b64(
      (GLOBAL_AS i32x2*)g, (LDS_AS i32x2*)l, 0, 0);
#else
  *l = *(const float2*)g;
#endif
}
__device__ __forceinline__ void async_b32(const float* g, float* l) {
#if __has_builtin(__builtin_amdgcn_global_load_async_to_lds_b32)
  __builtin_amdgcn_global_load_async_to_lds_b32(
      (GLOBAL_AS int*)g, (LDS_AS int*)l, 0, 0);
#else
  *l = *g;
#endif
}
__device__ __forceinline__ void async_wait_all() {
#if __has_builtin(__builtin_amdgcn_s_wait_asynccnt)
  __builtin_amdgcn_s_wait_asynccnt(0);
#else
  asm volatile("s_wait_asynccnt 0x0" ::: "memory");
#endif
}

// ---------------------------------------------------------------------------
// Round-robin tournament pairing for parallel Jacobi: round r, slot s -> (p,q)
// ---------------------------------------------------------------------------
__device__ __forceinline__ void rr_pair(int r, int s, int n, int& p, int& q) {
  int a = (s == 0) ? 0 : (1 + (s - 1 + r) % (n - 1));
  int b = 1 + (n - 2 - s + r) % (n - 1);
  p = a < b ? a : b;
  q = a < b ? b : a;
}

// ---------------------------------------------------------------------------
// Persistent one-workgroup cyclic Jacobi eigensolver for symmetric 1024x1024.
// ---------------------------------------------------------------------------
__global__ __launch_bounds__(1024) void jacobi_eigh_kernel(float* __restrict__ Awk,
                                                           float* __restrict__ V) {
  __shared__ float sc[LR / 2];
  __shared__ float ss[LR / 2];
  const int tid = threadIdx.x;
  for (int sweep = 0; sweep < JSWEEPS; ++sweep) {
    for (int r = 0; r < LR - 1; ++r) {
      if (tid < LR / 2) {
        int p, q;
        rr_pair(r, tid, LR, p, q);
        float app = Awk[(size_t)p * LR + p];
        float aqq = Awk[(size_t)q * LR + q];
        float apq = Awk[(size_t)p * LR + q];
        float c = 1.0f, sn = 0.0f;
        if (fabsf(apq) > 1e-30f) {
          float theta = 0.5f * atan2f(2.0f * apq, app - aqq);
          sincosf(theta, &sn, &c);
        }
        sc[tid] = c;
        ss[tid] = sn;
      }
      __syncthreads();
      {  // A <- A*J and V <- V*J; thread owns row tid
        float* Ar = Awk + (size_t)tid * LR;
        float* Vr = V + (size_t)tid * LR;
        for (int s = 0; s < LR / 2; ++s) {
          int p, q;
          rr_pair(r, s, LR, p, q);
          const float c = sc[s], sn = ss[s];
          float ap = Ar[p], aq = Ar[q];
          Ar[p] = c * ap - sn * aq;
          Ar[q] = sn * ap + c * aq;
          float vp = Vr[p], vq = Vr[q];
          Vr[p] = c * vp - sn * vq;
          Vr[q] = sn * vp + c * vq;
        }
      }
      __syncthreads();
      {  // A <- J^T*A; thread owns column tid
        for (int s = 0; s < LR / 2; ++s) {
          int p, q;
          rr_pair(r, s, LR, p, q);
          const float c = sc[s], sn = ss[s];
          float ap = Awk[(size_t)p * LR + tid];
          float aq = Awk[(size_t)q * LR + tid];
          Awk[(size_t)p * LR + tid] = c * ap - sn * aq;
          Awk[(size_t)q * LR + tid] = sn * ap + c * aq;
        }
      }
      __syncthreads();
    }
  }
}

__global__ __launch_bounds__(1024) void sort_eigs_kernel(const float* __restrict__ Awk,
                                                         int* __restrict__ idx_out) {
  __shared__ float vals[LR];
  __shared__ int idx[LR];
  const int t = threadIdx.x;
  vals[t] = Awk[(size_t)t * LR + t];
  idx[t] = t;
  __syncthreads();
  for (int phase = 0; phase < LR; ++phase) {
    int ii = (phase & 1) + 2 * t;
    if (t < LR / 2 && ii + 1 < LR) {
      if (vals[ii] > vals[ii + 1]) {
        float tv = vals[ii]; vals[ii] = vals[ii + 1]; vals[ii + 1] = tv;
        int ti = idx[ii]; idx[ii] = idx[ii + 1]; idx[ii + 1] = ti;
      }
    }
    __syncthreads();
  }
  idx_out[t] = idx[t];
}

__global__ void permute_cols_kernel(const float* __restrict__ V,
                                    const int* __restrict__ idx,
                                    float* __restrict__ Vsort) {
  int i = blockIdx.x * 256 + threadIdx.x;
  if (i < LR * LR) {
    int row = i / LR, col = i % LR;
    Vsort[i] = V[(size_t)row * LR + idx[col]];
  }
}

__global__ void init_eig_kernel(const float* __restrict__ A,
                                float* __restrict__ Acur,
                                float* __restrict__ V) {
  int i = blockIdx.x * 256 + threadIdx.x;
  if (i < LR * LR) {
    Acur[i] = A[i];
    V[i] = ((i / LR) == (i % LR)) ? 1.0f : 0.0f;
  }
}

__global__ void prep_ws_kernel(const float* __restrict__ W, float* __restrict__ Ws) {
  int i = blockIdx.x * 256 + threadIdx.x;
  if (i < HR * LR) {
    int r = i / LR, c = i % LR;
    Ws[i] = W[(size_t)r * HR + c] + W[(size_t)r * HR + LR + c];
  }
}

__global__ void softmax8_kernel(const float* __restrict__ alphas, float* __restrict__ out) {
  float m = -1e30f;
  for (int i = 0; i < KCOMM; ++i) m = fmaxf(m, alphas[i]);
  float e[KCOMM], sum = 0.0f;
  for (int i = 0; i < KCOMM; ++i) { e[i] = __expf(alphas[i] - m); sum += e[i]; }
  for (int i = 0; i < KCOMM; ++i) out[i] = e[i] / sum;
}

// ---------------------------------------------------------------------------
// Tiled f32 WMMA GEMM with double-buffered async LDS staging.
//   Block: 256 threads (8 waves), 128x128 output tile, K-slab 32.
//   Wave:  32x64 subtile = 2x4 grid of 16x16 WMMA accumulators.
//   LDS:   float2[kp][m] fragment layout; row stride 144 float2 (1152 B) so
//          the kp+1 row read by lanes 16-31 lands 32 banks away from lanes
//          0-15 -> conflict-free 32-lane x 8 B ds_load.
//   TA/TB: operand transpose; EPI==1: store (row==col) ? 1 : |v|.
// ---------------------------------------------------------------------------
#define LDSTRIDE 144

template <int TA, int TB, int EPI>
__global__ __launch_bounds__(256) void wmma_gemm_tiled(
    const float* __restrict__ A, const float* __restrict__ B, float* __restrict__ C,
    int K, int lda, int ldb, int ldc) {
  __shared__ float2 sA[2][16][LDSTRIDE];
  __shared__ float2 sB[2][16][LDSTRIDE];

  const int tid = threadIdx.x;
  const int wave = tid >> 5;
  const int lane = tid & 31;
  const int hv = lane >> 4;   // lane half -> K pair select
  const int ln = lane & 15;
  const int bm0 = blockIdx.y * 128;
  const int bn0 = blockIdx.x * 128;
  const int wm0 = (wave & 3) * 32;   // wave tile origin within block
  const int wn0 = (wave >> 2) * 64;

  v8f acc[2][4] = {};

  // stage K-slab [kb, kb+32) of A (128 rows) and B (128 cols) into buffer buf
  auto stage = [&](int buf, int kb) {
#pragma unroll
    for (int i = 0; i < 8; ++i) {
      int e = tid + (i << 8);         // 0..2047
      int kp = e >> 7;                // K-pair 0..15
      int m = e & 127;
      int gk = kb + (kp << 1);
      if (TA == 0) {
        async_b64(&A[(size_t)(bm0 + m) * lda + gk], &sA[buf][kp][m]);
      } else {
        async_b32(&A[(size_t)gk * lda + bm0 + m], &sA[buf][kp][m].x);
        async_b32(&A[(size_t)(gk + 1) * lda + bm0 + m], &sA[buf][kp][m].y);
      }
      if (TB == 0) {
        async_b32(&B[(size_t)gk * ldb + bn0 + m], &sB[buf][kp][m].x);
        async_b32(&B[(size_t)(gk + 1) * ldb + bn0 + m], &sB[buf][kp][m].y);
      } else {
        async_b64(&B[(size_t)(bn0 + m) * ldb + gk], &sB[buf][kp][m]);
      }
    }
  };

  stage(0, 0);
  async_wait_all();
  __syncthreads();

  int cur = 0;
  for (int kb = 0; kb < K; kb += 32) {
    if (kb + 32 < K) stage(cur ^ 1, kb + 32);   // async fill of next buffer
#pragma unroll
    for (int ks = 0; ks < 32; ks += 4) {
      const int kp = (ks >> 1) + hv;
      v2f a[2], b[4];
#pragma unroll
      for (int i = 0; i < 2; ++i) {
        float2 t = sA[cur][kp][wm0 + i * 16 + ln];
        a[i].x = t.x; a[i].y = t.y;
      }
#pragma unroll
      for (int j = 0; j < 4; ++j) {
        float2 t = sB[cur][kp][wn0 + j * 16 + ln];
        b[j].x = t.x; b[j].y = t.y;
      }
#pragma unroll
      for (int i = 0; i < 2; ++i)
#pragma unroll
        for (int j = 0; j < 4; ++j)
          acc[i][j] = __builtin_amdgcn_wmma_f32_16x16x4_f32(
              false, a[i], false, b[j], (short)0, acc[i][j], false, false);
    }
    async_wait_all();
    __syncthreads();
    cur ^= 1;
  }

#pragma unroll
  for (int i = 0; i < 2; ++i)
#pragma unroll
    for (int j = 0; j < 4; ++j)
#pragma unroll
      for (int r = 0; r < 8; ++r) {
        const int row = bm0 + wm0 + i * 16 + r + (hv << 3);
        const int col = bn0 + wn0 + j * 16 + ln;
        float v = acc[i][j][r];
        if (EPI == 1) v = (row == col) ? 1.0f : fabsf(v);
        C[(size_t)row * ldc + col] = v;
      }
}

// ---------------------------------------------------------------------------
// t_k = (adj .* mask_k) @ U_k   (N = RANK = 16). HBM-bound mask stream.
// 256-thread blocks; each wave owns one 16-row M-tile.
// ---------------------------------------------------------------------------
__global__ __launch_bounds__(256) void masked_lowrank_t_kernel(
    const float* __restrict__ adj, const float* __restrict__ masks,
    const float* __restrict__ Us, float* __restrict__ tK) {
  const int wave = threadIdx.x >> 5;
  const int tm = blockIdx.x * 128 + wave * 16;
  const int kc = blockIdx.y;
  const int lane = threadIdx.x & 31;
  const int hv = lane >> 4;
  const int ln = lane & 15;
  const int kb = hv << 1;
  const float* mk = masks + (size_t)kc * HR * HR;
  const float* Uk = Us + (size_t)kc * HR * RANK;
  const size_t rowOff = (size_t)(tm + ln) * HR;

  v8f acc = {};
  for (int kk = 0; kk < HR; kk += 4) {
    const int k0 = kk + kb;
    if ((kk & 127) == 0 && kk + 512 < HR) {
      __builtin_prefetch(&mk[rowOff + kk + 512], 0, 0);
    }
    v2f av, bv;
    av.x = adj[rowOff + k0] * mk[rowOff + k0];
    av.y = adj[rowOff + k0 + 1] * mk[rowOff + k0 + 1];
    bv.x = Uk[(size_t)k0 * RANK + ln];
    bv.y = Uk[(size_t)(k0 + 1) * RANK + ln];
    acc = __builtin_amdgcn_wmma_f32_16x16x4_f32(false, av, false, bv,
                                                (short)0, acc, false, false);
  }
#pragma unroll
  for (int r = 0; r < 8; ++r) {
    tK[(size_t)kc * HR * RANK + (size_t)(tm + r + (hv << 3)) * RANK + ln] = acc[r];
  }
}

// ---------------------------------------------------------------------------
// out_adj = adj + sum_k alpha_k * mask_k .* (t_k @ Vs_k^T)
// ---------------------------------------------------------------------------
__global__ __launch_bounds__(256) void apply_corr_kernel(
    const float* __restrict__ adj, const float* __restrict__ tK,
    const float* __restrict__ Vs, const float* __restrict__ masks,
    const float* __restrict__ alpha_s, float* __restrict__ out_adj) {
  __shared__ float sT[16][16];
  __shared__ float sV[16][16];
  const int tx = threadIdx.x, ty = threadIdx.y;
  const int j0 = blockIdx.x << 4;
  const int i0 = blockIdx.y << 4;
  const int i = i0 + ty, j = j0 + tx;
  float acc = 0.0f;
  for (int kc = 0; kc < KCOMM; ++kc) {
    sT[ty][tx] = tK[(size_t)kc * HR * RANK + (size_t)(i0 + ty) * RANK + tx];
    sV[ty][tx] = Vs[(size_t)kc * HR * RANK + (size_t)(j0 + ty) * RANK + tx];
    __syncthreads();
    float d = 0.0f;
#pragma unroll
    for (int r = 0; r < RANK; ++r) d += sT[ty][r] * sV[tx][r];
    acc += alpha_s[kc] * masks[(size_t)kc * HR * HR + (size_t)i * HR + j] * d;
    __syncthreads();
  }
  out_adj[(size_t)i * HR + j] = adj[(size_t)i * HR + j] + acc;
}

// ---------------------------------------------------------------------------
extern "C" void kernel_launch(void* const* d_in, const int* in_sizes, int n_in,
                              void* d_out, int out_size, void* d_ws, size_t ws_size,
                              hipStream_t stream) {
  const float* A      = (const float*)d_in[0];  // (LR, LR)
  const float* X      = (const float*)d_in[1];  // (LR, HR)
  const float* W      = (const float*)d_in[2];  // (HR, HR)
  const float* Us     = (const float*)d_in[3];  // (K, HR, RANK)
  const float* Vs     = (const float*)d_in[4];  // (K, HR, RANK)
  const float* alphas = (const float*)d_in[5];  // (K,)
  const float* masks  = (const float*)d_in[6];  // (K, HR, HR)

  float* out_adj = (float*)d_out;                    // (HR, HR)
  float* xout    = (float*)d_out + (size_t)HR * HR;  // (HR, HR)

  float* ws      = (float*)d_ws;
  float* Acur    = ws;                               // LR*LR (reused for Vsort)
  float* V       = Acur + (size_t)LR * LR;           // LR*LR
  float* Ws      = V + (size_t)LR * LR;              // HR*LR
  float* T       = Ws + (size_t)HR * LR;             // LR*HR
  float* adj     = T + (size_t)LR * HR;              // HR*HR
  float* tK      = adj + (size_t)HR * HR;            // K*HR*RANK
  float* alpha_s = tK + (size_t)KCOMM * HR * RANK;   // K (pad 16)
  int*   idx     = (int*)(alpha_s + 16);             // LR
  float* Vsort   = Acur;                             // safe reuse after sort

  init_eig_kernel<<<(LR * LR + 255) / 256, 256, 0, stream>>>(A, Acur, V);
  prep_ws_kernel<<<(HR * LR + 255) / 256, 256, 0, stream>>>(W, Ws);
  softmax8_kernel<<<1, 1, 0, stream>>>(alphas, alpha_s);

  jacobi_eigh_kernel<<<1, 1024, 0, stream>>>(Acur, V);
  sort_eigs_kernel<<<1, 1024, 0, stream>>>(Acur, idx);
  permute_cols_kernel<<<(LR * LR + 255) / 256, 256, 0, stream>>>(V, idx, Vsort);

  // T = U^T X : (LR x HR), A transposed
  wmma_gemm_tiled<1, 0, 0><<<dim3(HR / 128, LR / 128), 256, 0, stream>>>(
      Vsort, X, T, LR, LR, HR, HR);
  // adj = |Ws @ T| with unit diagonal : (HR x HR)
  wmma_gemm_tiled<0, 0, 1><<<dim3(HR / 128, HR / 128), 256, 0, stream>>>(
      Ws, T, adj, LR, LR, HR, HR);

  // low-rank community corrections
  masked_lowrank_t_kernel<<<dim3(HR / 128, KCOMM), 256, 0, stream>>>(adj, masks, Us, tK);
  apply_corr_kernel<<<dim3(HR / 16, HR / 16), dim3(16, 16), 0, stream>>>(
      adj, tK, Vs, masks, alpha_s, out_adj);

  // X_out = |G G^T| with unit diagonal (G G^T already symmetric)
  wmma_gemm_tiled<0, 1, 1><<<dim3(HR / 128, HR / 128), 256, 0, stream>>>(
      out_adj, out_adj, xout, HR, HR, HR, HR);
}